// EdgeAwareGAT_79645873537110
// MI455X (gfx1250) — compile-verified
//
#include <hip/hip_runtime.h>
#include <hip/hip_bf16.h>
#include <math.h>

typedef __attribute__((ext_vector_type(2))) float v2f;
typedef __attribute__((ext_vector_type(8))) float v8f;

#define NEG_SLOPE 0.2f

// ---------------------------------------------------------------- utilities

__global__ void fill_f32(float* __restrict__ p, float v, long n) {
    long i = (long)blockIdx.x * blockDim.x + threadIdx.x;
    if (i < n) p[i] = v;
}

__device__ __forceinline__ void atomicMaxF(float* addr, float val) {
    unsigned int* ua = (unsigned int*)addr;
    unsigned int old = *ua;
    while (true) {
        float f = __uint_as_float(old);
        if (f >= val) break;
        unsigned int assumed = old;
        old = atomicCAS(ua, assumed, __float_as_uint(val));
        if (old == assumed) break;
    }
}

// ------------------------------------------------- self-loop edge attributes

__global__ void loop_accum(const long long* __restrict__ dst,
                           const float* __restrict__ eattr,
                           float* __restrict__ loop_sum, float* __restrict__ deg,
                           int E) {
    int e = blockIdx.x * blockDim.x + threadIdx.x;
    if (e >= E) return;
    int d = (int)dst[e];
    atomicAdd(&deg[d], 1.0f);
    const float* ea = &eattr[(long)e * 16];
    float* ls = &loop_sum[(long)d * 16];
#pragma unroll
    for (int k = 0; k < 16; k++) atomicAdd(&ls[k], ea[k]);
}

__global__ void loop_final(float* __restrict__ loop_attr,
                           const float* __restrict__ deg, long n16) {
    long i = (long)blockIdx.x * blockDim.x + threadIdx.x;
    if (i >= n16) return;
    float d = deg[i >> 4];
    loop_attr[i] = loop_attr[i] / fmaxf(d, 1.0f);
}

// ------------------------------------------------------- fp32 WMMA GEMM
// C[M,N] = A[M,K] @ B[K,N] + bias[N]   (one wave computes a 16x16 tile)
// A-frag (16x4 f32): lanes 0-15 -> M=lane, VGPR0=K0,VGPR1=K1; lanes 16-31 -> K2,K3
// B-frag (4x16 f32): lanes hold N=lane%16; VGPR0=K(half*2), VGPR1=K(half*2+1)
// C/D: VGPR j, lanes 0-15 -> row j, lanes 16-31 -> row j+8; col = lane%16

__global__ void wmma_gemm_f32(const float* __restrict__ A,
                              const float* __restrict__ B,
                              const float* __restrict__ bias,
                              float* __restrict__ C,
                              int M, int K, int N) {
    int m0 = blockIdx.x * 16;
    int n0 = blockIdx.y * 16;
    int l = threadIdx.x;          // 0..31
    int half = l >> 4;
    int lm = l & 15;
    int arow = m0 + lm;
    if (arow >= M) arow = M - 1;  // clamp (keep EXEC full)
    const float* Ar = &A[(long)arow * K];
    v8f acc = {};
    for (int k0 = 0; k0 < K; k0 += 4) {
        int ka = k0 + half * 2;
        v2f a;  a.x = Ar[ka];                 a.y = Ar[ka + 1];
        v2f b;  b.x = B[(long)ka * N + n0 + lm];
                b.y = B[(long)(ka + 1) * N + n0 + lm];
        acc = __builtin_amdgcn_wmma_f32_16x16x4_f32(
                  false, a, false, b, (short)0, acc, false, false);
    }
#pragma unroll
    for (int j = 0; j < 8; j++) {
        int row = m0 + j + half * 8;
        if (row < M) {
            int col = n0 + lm;
            C[(long)row * N + col] = acc[j] + bias[col];
        }
    }
}

// ------------------------------------------------- GATv2 attention passes
// one thread per (edge2, head); edges e2 >= E are the self loops.

__global__ void gat_alpha(const long long* __restrict__ src,
                          const long long* __restrict__ dst,
                          const float* __restrict__ eattr,
                          const float* __restrict__ loop_attr,
                          const float* __restrict__ xl,
                          const float* __restrict__ xr,
                          const float* __restrict__ We,   // [16, H*C]
                          const float* __restrict__ att,  // [H, C]
                          float* __restrict__ alphaE,     // [E2, H]
                          float* __restrict__ amax,       // [N, H]
                          int E, int Nn, int H, int C) {
    long idx = (long)blockIdx.x * blockDim.x + threadIdx.x;
    long tot = (long)(E + Nn) * H;
    if (idx >= tot) return;
    int e2 = (int)(idx / H);
    int h  = (int)(idx % H);
    int s, d;
    const float* eap;
    if (e2 < E) { s = (int)src[e2]; d = (int)dst[e2]; eap = &eattr[(long)e2 * 16]; }
    else        { s = d = e2 - E;                     eap = &loop_attr[(long)(e2 - E) * 16]; }
    float eat[16];
#pragma unroll
    for (int k = 0; k < 16; k++) eat[k] = eap[k];
    const int HC = H * C;
    const float* xls  = &xl[(long)s * HC + h * C];
    const float* xrd  = &xr[(long)d * HC + h * C];
    const float* Wec  = &We[h * C];
    const float* atth = &att[h * C];
    float acc = 0.0f;
    for (int c = 0; c < C; c++) {
        float ea = 0.0f;
#pragma unroll
        for (int k = 0; k < 16; k++) ea += eat[k] * Wec[k * HC + c];
        float m = xls[c] + xrd[c] + ea;
        m = (m > 0.0f) ? m : NEG_SLOPE * m;
        acc += m * atth[c];
    }
    alphaE[idx] = acc;
    atomicMaxF(&amax[(long)d * H + h], acc);
}

__global__ void gat_exp(const long long* __restrict__ dst,
                        float* __restrict__ alphaE,
                        const float* __restrict__ amax,
                        float* __restrict__ denom,
                        int E, int Nn, int H) {
    long idx = (long)blockIdx.x * blockDim.x + threadIdx.x;
    long tot = (long)(E + Nn) * H;
    if (idx >= tot) return;
    int e2 = (int)(idx / H);
    int h  = (int)(idx % H);
    int d = (e2 < E) ? (int)dst[e2] : (e2 - E);
    float a = expf(alphaE[idx] - amax[(long)d * H + h]);
    alphaE[idx] = a;
    atomicAdd(&denom[(long)d * H + h], a);
}

__global__ void gat_agg(const long long* __restrict__ src,
                        const long long* __restrict__ dst,
                        const float* __restrict__ xl,
                        const float* __restrict__ alphaE,
                        const float* __restrict__ denom,
                        float* __restrict__ agg,
                        int E, int Nn, int H, int C) {
    long idx = (long)blockIdx.x * blockDim.x + threadIdx.x;
    long tot = (long)(E + Nn) * H;
    if (idx >= tot) return;
    int e2 = (int)(idx / H);
    int h  = (int)(idx % H);
    int s, d;
    if (e2 < E) { s = (int)src[e2]; d = (int)dst[e2]; }
    else        { s = d = e2 - E; }
    const int HC = H * C;
    float w = alphaE[idx] / denom[(long)d * H + h];
    const float* xs = &xl[(long)s * HC + h * C];
    float* ag = &agg[(long)d * HC + h * C];
    for (int c = 0; c < C; c++) atomicAdd(&ag[c], xs[c] * w);
}

__global__ void bias_elu(float* __restrict__ buf, const float* __restrict__ bias,
                         int HC, long n) {
    long i = (long)blockIdx.x * blockDim.x + threadIdx.x;
    if (i >= n) return;
    float v = buf[i] + bias[(int)(i % HC)];
    buf[i] = (v > 0.0f) ? v : (expf(v) - 1.0f);
}

// ------------------------------------------------- edge classifier (WMMA)
// logits[e] = bc2 + sum_j relu( feats[e]·Wc1[:,j] + bc1[j] ) * Wc2[j]
// feats = [h2[src](32) | h2[dst](32) | edge_attr(16)]  -> K=80, Nh=32
// 8 waves/block; each wave gathers a 16x80 A-tile into LDS, runs 20 k-steps
// of v_wmma_f32_16x16x4_f32 on two 16x16 N-tiles, then reduces with shuffles.

#define CL_TPB 8

__global__ void edge_classifier(const long long* __restrict__ src,
                                const long long* __restrict__ dst,
                                const float* __restrict__ h2,
                                const float* __restrict__ eattr,
                                const float* __restrict__ Wc1,  // [80,32]
                                const float* __restrict__ bc1,  // [32]
                                const float* __restrict__ Wc2,  // [32]
                                const float* __restrict__ bc2,  // [1]
                                float* __restrict__ out, int E) {
    __shared__ float sW[80 * 32];
    __shared__ float sb1[32];
    __shared__ float sw2[32];
    __shared__ float sA[CL_TPB][16 * 80];

    int tid  = threadIdx.x;      // 0..255
    int wave = tid >> 5;
    int lane = tid & 31;

    for (int i = tid; i < 80 * 32; i += 256) sW[i] = Wc1[i];
    if (tid < 32) { sb1[tid] = bc1[tid]; sw2[tid] = Wc2[tid]; }

    long tile = (long)blockIdx.x * CL_TPB + wave;
    long base = tile * 16;

    // gather 16 rows x 80 cols of feats into LDS (clamped, store-guarded later)
    for (int i = lane; i < 16 * 80; i += 32) {
        int r = i / 80, c = i % 80;
        long e = base + r;
        if (e >= E) e = (long)E - 1;
        float v;
        if (c < 32)      v = h2[(long)src[e] * 32 + c];
        else if (c < 64) v = h2[(long)dst[e] * 32 + (c - 32)];
        else             v = eattr[e * 16 + (c - 64)];
        sA[wave][i] = v;
    }
    __syncthreads();

    int half = lane >> 4, lm = lane & 15;
    const float* Ar = &sA[wave][0];
    v8f c0 = {}, c1 = {};
    for (int k0 = 0; k0 < 80; k0 += 4) {
        int ka = k0 + half * 2;
        v2f a;   a.x  = Ar[lm * 80 + ka];       a.y  = Ar[lm * 80 + ka + 1];
        v2f b0;  b0.x = sW[ka * 32 + lm];       b0.y = sW[(ka + 1) * 32 + lm];
        v2f b1;  b1.x = sW[ka * 32 + 16 + lm];  b1.y = sW[(ka + 1) * 32 + 16 + lm];
        c0 = __builtin_amdgcn_wmma_f32_16x16x4_f32(false, a, false, b0, (short)0, c0, false, false);
        c1 = __builtin_amdgcn_wmma_f32_16x16x4_f32(false, a, false, b1, (short)0, c1, false, false);
    }

    float b1a = sb1[lm],      b1b = sb1[16 + lm];
    float w2a = sw2[lm],      w2b = sw2[16 + lm];
    float bo  = bc2[0];
#pragma unroll
    for (int j = 0; j < 8; j++) {
        float p = fmaxf(c0[j] + b1a, 0.0f) * w2a +
                  fmaxf(c1[j] + b1b, 0.0f) * w2b;
        p += __shfl_xor(p, 1, 16);
        p += __shfl_xor(p, 2, 16);
        p += __shfl_xor(p, 4, 16);
        p += __shfl_xor(p, 8, 16);
        if (lm == 0) {
            long e = base + j + half * 8;
            if (e < E) out[e] = p + bo;
        }
    }
}

// ---------------------------------------------------------------- driver

extern "C" void kernel_launch(void* const* d_in, const int* in_sizes, int n_in,
                              void* d_out, int out_size, void* d_ws, size_t ws_size,
                              hipStream_t stream) {
    const float*      x     = (const float*)d_in[0];
    const long long*  eidx  = (const long long*)d_in[1];
    const float*      eattr = (const float*)d_in[2];
    const float* W1l  = (const float*)d_in[3];  const float* b1l  = (const float*)d_in[4];
    const float* W1r  = (const float*)d_in[5];  const float* b1r  = (const float*)d_in[6];
    const float* W1e  = (const float*)d_in[7];  const float* att1 = (const float*)d_in[8];
    const float* bias1= (const float*)d_in[9];
    const float* W2l  = (const float*)d_in[10]; const float* b2l  = (const float*)d_in[11];
    const float* W2r  = (const float*)d_in[12]; const float* b2r  = (const float*)d_in[13];
    const float* W2e  = (const float*)d_in[14]; const float* att2 = (const float*)d_in[15];
    const float* bias2= (const float*)d_in[16];
    const float* Wc1  = (const float*)d_in[17]; const float* bc1  = (const float*)d_in[18];
    const float* Wc2  = (const float*)d_in[19]; const float* bc2  = (const float*)d_in[20];
    float* out = (float*)d_out;

    const int Nn = in_sizes[0] / 64;     // 50000
    const int E  = in_sizes[2] / 16;     // 1000000
    const long long* src = eidx;
    const long long* dst = eidx + E;
    const int E2 = E + Nn;

    // workspace layout (floats)
    float* W = (float*)d_ws;
    size_t off = 0;
    auto alloc = [&](size_t n) { float* p = W + off; off += n; return p; };
    float* loop_attr = alloc((size_t)Nn * 16);
    float* deg       = alloc((size_t)Nn);
    float* xl1       = alloc((size_t)Nn * 128);
    float* xr1       = alloc((size_t)Nn * 128);
    float* agg1      = alloc((size_t)Nn * 128);   // -> h1 in place
    float* amax1     = alloc((size_t)Nn * 4);
    float* denom1    = alloc((size_t)Nn * 4);
    float* alphaE1   = alloc((size_t)E2 * 4);
    float* xl2       = alloc((size_t)Nn * 32);
    float* xr2       = alloc((size_t)Nn * 32);
    float* agg2      = alloc((size_t)Nn * 32);    // -> h2 in place
    float* amax2     = alloc((size_t)Nn);
    float* denom2    = alloc((size_t)Nn);
    float* alphaE2   = alloc((size_t)E2);
    (void)ws_size;

    auto cdiv = [](long a, long b) { return (unsigned int)((a + b - 1) / b); };
    const int T = 256;

    // init accumulators
    fill_f32<<<cdiv((long)Nn * 16, T), T, 0, stream>>>(loop_attr, 0.0f, (long)Nn * 16);
    fill_f32<<<cdiv(Nn, T),          T, 0, stream>>>(deg,    0.0f, Nn);
    fill_f32<<<cdiv((long)Nn * 128,T),T, 0, stream>>>(agg1,  0.0f, (long)Nn * 128);
    fill_f32<<<cdiv((long)Nn * 4,  T),T, 0, stream>>>(amax1, -3.0e38f, (long)Nn * 4);
    fill_f32<<<cdiv((long)Nn * 4,  T),T, 0, stream>>>(denom1, 0.0f, (long)Nn * 4);
    fill_f32<<<cdiv((long)Nn * 32, T),T, 0, stream>>>(agg2,  0.0f, (long)Nn * 32);
    fill_f32<<<cdiv(Nn, T),          T, 0, stream>>>(amax2, -3.0e38f, Nn);
    fill_f32<<<cdiv(Nn, T),          T, 0, stream>>>(denom2, 0.0f, Nn);

    // self-loop edge attrs (segment mean over dst)
    loop_accum<<<cdiv(E, T), T, 0, stream>>>(dst, eattr, loop_attr, deg, E);
    loop_final<<<cdiv((long)Nn * 16, T), T, 0, stream>>>(loop_attr, deg, (long)Nn * 16);

    // layer-1 node transforms (WMMA f32)
    wmma_gemm_f32<<<dim3(cdiv(Nn, 16), 8), 32, 0, stream>>>(x, W1l, b1l, xl1, Nn, 64, 128);
    wmma_gemm_f32<<<dim3(cdiv(Nn, 16), 8), 32, 0, stream>>>(x, W1r, b1r, xr1, Nn, 64, 128);

    // layer-1 attention (H=4, C=32)
    {
        long tot = (long)E2 * 4;
        gat_alpha<<<cdiv(tot, T), T, 0, stream>>>(src, dst, eattr, loop_attr, xl1, xr1,
                                                  W1e, att1, alphaE1, amax1, E, Nn, 4, 32);
        gat_exp<<<cdiv(tot, T), T, 0, stream>>>(dst, alphaE1, amax1, denom1, E, Nn, 4);
        gat_agg<<<cdiv(tot, T), T, 0, stream>>>(src, dst, xl1, alphaE1, denom1, agg1,
                                                E, Nn, 4, 32);
        bias_elu<<<cdiv((long)Nn * 128, T), T, 0, stream>>>(agg1, bias1, 128, (long)Nn * 128);
    }
    float* h1 = agg1;

    // layer-2 node transforms (WMMA f32)
    wmma_gemm_f32<<<dim3(cdiv(Nn, 16), 2), 32, 0, stream>>>(h1, W2l, b2l, xl2, Nn, 128, 32);
    wmma_gemm_f32<<<dim3(cdiv(Nn, 16), 2), 32, 0, stream>>>(h1, W2r, b2r, xr2, Nn, 128, 32);

    // layer-2 attention (H=1, C=32)
    {
        long tot = (long)E2;
        gat_alpha<<<cdiv(tot, T), T, 0, stream>>>(src, dst, eattr, loop_attr, xl2, xr2,
                                                  W2e, att2, alphaE2, amax2, E, Nn, 1, 32);
        gat_exp<<<cdiv(tot, T), T, 0, stream>>>(dst, alphaE2, amax2, denom2, E, Nn, 1);
        gat_agg<<<cdiv(tot, T), T, 0, stream>>>(src, dst, xl2, alphaE2, denom2, agg2,
                                                E, Nn, 1, 32);
        bias_elu<<<cdiv((long)Nn * 32, T), T, 0, stream>>>(agg2, bias2, 32, (long)Nn * 32);
    }
    float* h2 = agg2;

    // edge classifier (gathered-A WMMA GEMM + fused relu/dot reduction)
    edge_classifier<<<cdiv((long)E, 16 * CL_TPB), 256, 0, stream>>>(
        src, dst, h2, eattr, Wc1, bc1, Wc2, bc2, out, E);
}